// SimpleGatedDeltaNet_11922829214182
// MI455X (gfx1250) — compile-verified
//
#include <hip/hip_runtime.h>
#include <hip/hip_bf16.h>

typedef __attribute__((ext_vector_type(2))) float v2f;
typedef __attribute__((ext_vector_type(8))) float v8f;

#define B_    4
#define T_    2048
#define C_    1024
#define H_    16
#define D_    64
#define M_TOT (B_ * T_)      // 8192 rows

// ---------------- fp32 WMMA projection GEMM -----------------
// Y[M,N] = X[M,K] @ W[K,N]; M=8192, K=N=1024. Uses V_WMMA_F32_16X16X4_F32.
// Block tile 128x128, BK=16, 256 threads = 8 waves in 4x2 grid; each wave owns
// a 32x64 strip (2 A-frags x 4 B-frags = 8 f32 accumulators -> 1.5 LDS
// loads/WMMA). Double-buffered LDS filled by CDNA5 async DMA
// (global_load_async_to_lds_b128, ASYNCcnt) -- no VGPR staging, no spills,
// DMA of tile it+1 overlaps WMMA compute of tile it.
#define BM  128
#define BN  128
#define BK  16
#define LDX (BK + 4)   // 20: row pitch 80B, 16B-aligned
#define LDW (BN + 4)   // 132: row pitch 528B, 16B-aligned

__device__ __forceinline__ void async_b128_to_lds(unsigned lds_addr, unsigned voff,
                                                  const float* sbase) {
    // GVS mode: mem_addr = SGPR64 + VGPR_I32 ; LDS dest = lds_addr (wave-relative)
    asm volatile("global_load_async_to_lds_b128 %0, %1, %2"
                 :: "v"(lds_addr), "v"(voff), "s"(sbase)
                 : "memory");
}
__device__ __forceinline__ void wait_async0() {
    asm volatile("s_wait_asynccnt 0x0" ::: "memory");
}

__global__ __launch_bounds__(256)
void proj_gemm_kernel(const float* __restrict__ X,
                      const float* __restrict__ W0, const float* __restrict__ W1,
                      const float* __restrict__ W2,
                      float* __restrict__ Y0, float* __restrict__ Y1,
                      float* __restrict__ Y2) {
    const int N = C_, K = C_;
    const int zn = blockIdx.z;
    const float* __restrict__ W = (zn == 0) ? W0 : ((zn == 1) ? W1 : W2);
    float* __restrict__ Y       = (zn == 0) ? Y0 : ((zn == 1) ? Y1 : Y2);

    const int n0 = blockIdx.x * BN;
    const int m0 = blockIdx.y * BM;

    __shared__ float Xs[2][BM * LDX];   // 2 x 128x20 floats
    __shared__ float Ws[2][BK * LDW];   // 2 x  16x132 floats

    const int tid  = threadIdx.x;
    const int wave = tid >> 5;
    const int lane = tid & 31;
    const int lr   = lane & 15;   // row within A-frag / col within B-frag
    const int half = lane >> 4;   // selects K pair {0,1} vs {2,3}
    const int wr   = wave & 3;    // wave row block (32 rows each)
    const int wc   = wave >> 2;   // wave col block (64 cols each)
    const int mw0  = wr * 32;
    const int nw0  = wc * 64;

    // DMA geometry: per thread 2 x b128 for X (rows xrow, xrow+64) and
    // 2 x b128 for W (rows wrow, wrow+8).
    const int xrow = tid >> 2;
    const int xc4  = (tid & 3) << 2;
    const int wrow = tid >> 5;
    const int wc4  = (tid & 31) << 2;

    const float* sX = X + (size_t)m0 * K;   // uniform 64-bit base (SGPR pair)
    const float* sW = W + n0;

    unsigned voffX[2], voffW[2], ldsX[2][2], ldsW[2][2];
    #pragma unroll
    for (int i = 0; i < 2; ++i) {
        voffX[i] = (unsigned)(((xrow + i * 64) * K + xc4) * 4);
        voffW[i] = (unsigned)(((wrow + i * 8) * N + wc4) * 4);
        #pragma unroll
        for (int bf = 0; bf < 2; ++bf) {
            ldsX[bf][i] = (unsigned)(uintptr_t)&Xs[bf][(xrow + i * 64) * LDX + xc4];
            ldsW[bf][i] = (unsigned)(uintptr_t)&Ws[bf][(wrow + i * 8) * LDW + wc4];
        }
    }

    auto issue = [&](int bf, int k0) {
        #pragma unroll
        for (int i = 0; i < 2; ++i) {
            async_b128_to_lds(ldsX[bf][i], voffX[i] + (unsigned)(k0 * 4), sX);
            async_b128_to_lds(ldsW[bf][i], voffW[i] + (unsigned)(k0 * N * 4), sW);
        }
    };

    v8f acc[2][4] = {};

    const int NT = K / BK;          // 64 K-tiles
    issue(0, 0);
    int cur = 0;

    for (int it = 0; it < NT; ++it) {
        wait_async0();                            // tile `it` DMA landed
        __syncthreads();                          // visible to all; buf^1 free
        if (it + 1 < NT) issue(cur ^ 1, (it + 1) * BK);   // overlap next DMA

        const float* xs = &Xs[cur][0];
        const float* ws = &Ws[cur][0];
        #pragma unroll
        for (int kk = 0; kk < BK; kk += 4) {
            // A frags (16x4 f32): lane lr holds M=lr; VGPR0=K{0|2}, VGPR1=K{1|3}
            v2f a0, a1;
            a0.x = xs[(mw0 + lr) * LDX + kk + 2 * half];
            a0.y = xs[(mw0 + lr) * LDX + kk + 2 * half + 1];
            a1.x = xs[(mw0 + 16 + lr) * LDX + kk + 2 * half];
            a1.y = xs[(mw0 + 16 + lr) * LDX + kk + 2 * half + 1];
            #pragma unroll
            for (int j = 0; j < 4; ++j) {
                // B frag (4x16 f32): lane lr holds N=lr; VGPR0=K{0|2}, VGPR1=K{1|3}
                v2f bv;
                bv.x = ws[(kk + 2 * half) * LDW + nw0 + j * 16 + lr];
                bv.y = ws[(kk + 2 * half + 1) * LDW + nw0 + j * 16 + lr];
                acc[0][j] = __builtin_amdgcn_wmma_f32_16x16x4_f32(
                    false, a0, false, bv, (short)0, acc[0][j], false, false);
                acc[1][j] = __builtin_amdgcn_wmma_f32_16x16x4_f32(
                    false, a1, false, bv, (short)0, acc[1][j], false, false);
            }
        }
        cur ^= 1;
    }

    // C/D layout: VGPR r -> (M = r + 8*half, N = lr)
    #pragma unroll
    for (int g = 0; g < 2; ++g) {
        #pragma unroll
        for (int j = 0; j < 4; ++j) {
            #pragma unroll
            for (int r = 0; r < 8; ++r) {
                int row = m0 + mw0 + g * 16 + r + 8 * half;
                int col = n0 + nw0 + j * 16 + lr;
                Y[(size_t)row * N + col] = acc[g][j][r];
            }
        }
    }
}

// ---------------- gate GEMV: alpha/beta = sigmoid(x @ Wa/Wb) ----------------
__global__ __launch_bounds__(256)
void gate_kernel(const float* __restrict__ X, const float* __restrict__ Wa,
                 const float* __restrict__ Wb, float* __restrict__ Al,
                 float* __restrict__ Be) {
    const int wave = threadIdx.x >> 5;
    const int lane = threadIdx.x & 31;
    const int m    = blockIdx.x * 8 + wave;
    const float* xr = X + (size_t)m * C_;
    float pa = 0.f, pb = 0.f;
    #pragma unroll 8
    for (int i = 0; i < C_ / 32; ++i) {
        float xv = xr[lane + 32 * i];
        pa += xv * Wa[lane + 32 * i];
        pb += xv * Wb[lane + 32 * i];
    }
    #pragma unroll
    for (int off = 16; off > 0; off >>= 1) {
        pa += __shfl_xor(pa, off, 32);
        pb += __shfl_xor(pb, off, 32);
    }
    if (lane == 0) {
        Al[m] = 1.f / (1.f + __expf(-pa));
        Be[m] = 1.f / (1.f + __expf(-pb));
    }
}

// ---------------- gated delta recurrence ----------------
// One WG (256 thr) per (b,h). Thread (v = tid>>2, k-seg = (tid&3)*16) owns
// S[v][seg..seg+15] in 16 VGPRs. 4-lane shfl_xor reductions; no LDS/barriers.
// Next timestep's k/q/v/a/b are register-prefetched to hide load latency.
__global__ __launch_bounds__(256)
void scan_kernel(const float* __restrict__ Q, const float* __restrict__ K,
                 const float* __restrict__ V, const float* __restrict__ A,
                 const float* __restrict__ Bt, float* __restrict__ Out) {
    const int bh  = blockIdx.x;          // 0..63
    const int b   = bh >> 4;
    const int h   = bh & 15;
    const int tid = threadIdx.x;
    const int v   = tid >> 2;            // 0..63  (row of state)
    const int seg = (tid & 3) << 4;      // 0,16,32,48 (k segment)

    const size_t base = (size_t)b * T_ * C_ + (size_t)h * D_;

    float s[16];
    #pragma unroll
    for (int i = 0; i < 16; ++i) s[i] = 0.f;

    float4 kc[4], qc[4];
    float  vc, ac, bc;
    {   // preload t = 0
        const float* kp = K + base + seg;
        const float* qp = Q + base + seg;
        #pragma unroll
        for (int i = 0; i < 4; ++i) {
            kc[i] = *(const float4*)(kp + 4 * i);
            qc[i] = *(const float4*)(qp + 4 * i);
        }
        vc = V[base + v];
        ac = A[(size_t)b * T_];
        bc = Bt[(size_t)b * T_];
    }

    for (int t = 0; t < T_; ++t) {
        // prefetch step t+1 (clamped; loads are L2-resident)
        const int tn = (t + 1 < T_) ? (t + 1) : t;
        const size_t base2 = base + (size_t)tn * C_;
        float4 kn[4], qn[4];
        {
            const float* kp = K + base2 + seg;
            const float* qp = Q + base2 + seg;
            #pragma unroll
            for (int i = 0; i < 4; ++i) {
                kn[i] = *(const float4*)(kp + 4 * i);
                qn[i] = *(const float4*)(qp + 4 * i);
            }
        }
        const float vn = V[base2 + v];
        const float an = A[(size_t)b * T_ + tn];
        const float bn = Bt[(size_t)b * T_ + tn];

        // sk[v] = sum_k S[v,k] * k_t[k]   (partial over 16, reduce over 4 lanes)
        float p = 0.f;
        #pragma unroll
        for (int i = 0; i < 4; ++i) {
            p += s[4 * i + 0] * kc[i].x + s[4 * i + 1] * kc[i].y +
                 s[4 * i + 2] * kc[i].z + s[4 * i + 3] * kc[i].w;
        }
        p += __shfl_xor(p, 1, 32);
        p += __shfl_xor(p, 2, 32);

        // S = a*S + (b*v - a*b*sk) * k^T ;  o[v] = sum_k q_t[k]*S[v,k]
        const float coef = bc * vc - ac * bc * p;
        float o = 0.f;
        #pragma unroll
        for (int i = 0; i < 4; ++i) {
            s[4 * i + 0] = ac * s[4 * i + 0] + coef * kc[i].x;  o += qc[i].x * s[4 * i + 0];
            s[4 * i + 1] = ac * s[4 * i + 1] + coef * kc[i].y;  o += qc[i].y * s[4 * i + 1];
            s[4 * i + 2] = ac * s[4 * i + 2] + coef * kc[i].z;  o += qc[i].z * s[4 * i + 2];
            s[4 * i + 3] = ac * s[4 * i + 3] + coef * kc[i].w;  o += qc[i].w * s[4 * i + 3];
        }
        o += __shfl_xor(o, 1, 32);
        o += __shfl_xor(o, 2, 32);
        if ((tid & 3) == 0)
            Out[base + (size_t)t * C_ + v] = o;

        #pragma unroll
        for (int i = 0; i < 4; ++i) { kc[i] = kn[i]; qc[i] = qn[i]; }
        vc = vn; ac = an; bc = bn;
    }
}

extern "C" void kernel_launch(void* const* d_in, const int* in_sizes, int n_in,
                              void* d_out, int out_size, void* d_ws, size_t ws_size,
                              hipStream_t stream) {
    (void)in_sizes; (void)n_in; (void)out_size; (void)ws_size;
    const float* x  = (const float*)d_in[0];
    const float* Wq = (const float*)d_in[1];
    const float* Wk = (const float*)d_in[2];
    const float* Wv = (const float*)d_in[3];
    const float* Wa = (const float*)d_in[4];
    const float* Wb = (const float*)d_in[5];
    float* out = (float*)d_out;

    float* ws = (float*)d_ws;
    float* Qp = ws;                                   // 8192*1024
    float* Kp = Qp + (size_t)M_TOT * C_;              // 8192*1024
    float* Vp = Kp + (size_t)M_TOT * C_;              // 8192*1024
    float* Al = Vp + (size_t)M_TOT * C_;              // 8192
    float* Be = Al + M_TOT;                           // 8192

    dim3 g1(C_ / BN, M_TOT / BM, 3);
    proj_gemm_kernel<<<g1, dim3(256), 0, stream>>>(x, Wq, Wk, Wv, Qp, Kp, Vp);
    gate_kernel<<<dim3(M_TOT / 8), dim3(256), 0, stream>>>(x, Wa, Wb, Al, Be);
    scan_kernel<<<dim3(B_ * H_), dim3(256), 0, stream>>>(Qp, Kp, Vp, Al, Be, out);
}